// mini2d_o3_5875515261439
// MI455X (gfx1250) — compile-verified
//
#include <hip/hip_runtime.h>
#include <math.h>

typedef __attribute__((ext_vector_type(2))) float v2f;
typedef __attribute__((ext_vector_type(8))) float v8f;

#define NROW 512
#define TOT  576
#define NCHN 224
#define DD   144
#define NSEG 37
#define ZTOT 3392
#define BLKS_PER_ROW 81   /* 144*144/256 float4s per row, exact */

// workspace layout (float offsets)
#define WS_CG    0        // 11 * 128
#define WS_INV   1408     // 512*224
#define WS_MEAN  116096   // 224
#define WS_VAR   116320   // 224
#define WS_M     116544   // 512*16  (per-row gate-combined 4x4 matrix)

struct Ptr6 { const float* p[6]; };

// ---- static O3 structure ----
__constant__ int c_NT[6]      = {64,32,48,32,32,16};
__constant__ int c_LT[6]      = {0,0,1,1,2,2};
__constant__ int c_BOFF[7]    = {0,64,96,240,336,496,576};
__constant__ int c_CHOFF[6]   = {0,64,96,144,176,208};
__constant__ int c_METAOUT[6] = {224,80,256,160,224,144};
__constant__ int c_ZBASE[6]   = {0,224,304,1072,1552,2672};

__constant__ int c_SEG_T1[NSEG]   = {0,0,0,0,0,0, 1,1,1,1,1, 2,2,2, 2,2,2, 2,2, 2,2, 3,3,3, 3,3, 3,3, 4,4,4, 4,4,4, 5,5,5};
__constant__ int c_SEG_T2[NSEG]   = {0,1,2,3,4,5, 1,2,3,4,5, 2,2,2, 3,3,3, 4,4, 5,5, 3,3,3, 4,4, 5,5, 4,4,4, 5,5,5, 5,5,5};
__constant__ int c_SEG_TOUT[NSEG] = {0,1,2,3,4,5, 0,3,2,5,4, 0,2,4, 1,3,5, 2,4, 3,5, 0,2,4, 3,5, 2,4, 0,2,4, 1,3,5, 0,2,4};
__constant__ int c_SEG_TYPE[NSEG] = {0,0,1,1,2,2, 0,1,1,2,2, 3,4,5, 3,4,5, 6,7, 6,7, 3,4,5, 6,7, 6,7, 8,9,10, 8,9,10, 8,9,10};
__constant__ int c_SEG_NC[NSEG]   = {64,32,48,32,32,16, 32,32,32,32,16, 48,48,48, 32,32,32, 32,32, 16,16, 32,32,32, 32,32, 16,16, 32,32,32, 16,16,16, 16,16,16};
__constant__ int c_SEG_CHOFF[NSEG]= {0,0,0,0,0,0, 64,32,48,16,32, 96,80,48, 32,64,48, 128,96, 96,80, 144,160,128, 112,96, 192,160, 176,208,176, 64,144,128, 208,240,208};
__constant__ int c_SEG_BASE[NSEG+1] = {0,64,96,240,336,496,576,608,704,800,960,1040,1088,1232,1472,1504,1600,1760,1856,2016,2064,2144,2176,2272,2432,2528,2688,2736,2816,2848,2944,3104,3120,3168,3248,3264,3312,3392};

__constant__ int c_TYPL[11][3] = {{0,0,0},{0,1,1},{0,2,2},{1,1,0},{1,1,1},{1,1,2},{1,2,1},{1,2,2},{2,2,0},{2,2,1},{2,2,2}};

// ---------------- CG coefficient init (Racah + real basis, on device) ----------------
__device__ inline double dfact(int n){
  const double f[9] = {1.,1.,2.,6.,24.,120.,720.,5040.,40320.};
  return f[n];
}
__device__ double cgc_dev(int j1,int m1,int j2,int m2,int j3,int m3){
  if (m1+m2 != m3) return 0.0;
  double pre = sqrt((double)(2*j3+1) * dfact(j1+j2-j3)*dfact(j1-j2+j3)*dfact(-j1+j2+j3)/dfact(j1+j2+j3+1));
  pre *= sqrt(dfact(j1+m1)*dfact(j1-m1)*dfact(j2+m2)*dfact(j2-m2)*dfact(j3+m3)*dfact(j3-m3));
  int k0 = 0;
  if (j2-j3-m1 > k0) k0 = j2-j3-m1;
  if (j1-j3+m2 > k0) k0 = j1-j3+m2;
  int k1 = j1+j2-j3;
  if (j1-m1 < k1) k1 = j1-m1;
  if (j2+m2 < k1) k1 = j2+m2;
  double s = 0.0;
  for (int k=k0;k<=k1;k++){
    double d = dfact(k)*dfact(j1+j2-j3-k)*dfact(j1-m1-k)*dfact(j2+m2-k)*dfact(j3-j2+m1+k)*dfact(j3-j1-m2+k);
    s += ((k&1) ? -1.0 : 1.0)/d;
  }
  return pre*s;
}
__device__ void buildU(int l, double Ure[5][5], double Uim[5][5]){
  for (int i=0;i<5;i++) for (int j=0;j<5;j++){ Ure[i][j]=0.0; Uim[i][j]=0.0; }
  Ure[l][l] = 1.0;
  const double s = 0.70710678118654752440;
  for (int m=1;m<=l;m++){
    double sgn = (m&1) ? -1.0 : 1.0;
    Ure[l+m][l+m] = sgn*s;
    Ure[l+m][l-m] = s;
    Uim[l-m][l-m] = s;
    Uim[l-m][l+m] = -sgn*s;
  }
}
__global__ void cg_init_kernel(float* __restrict__ cg){
  int tid = threadIdx.x;
  for (int tt=0; tt<11; ++tt){
    int l1=c_TYPL[tt][0], l2=c_TYPL[tt][1], l3=c_TYPL[tt][2];
    int d1=2*l1+1, d2=2*l2+1, d3=2*l3+1;
    int entries = d1*d2*d3;
    double U1re[5][5],U1im[5][5],U2re[5][5],U2im[5][5],U3re[5][5],U3im[5][5];
    buildU(l1,U1re,U1im); buildU(l2,U2re,U2im); buildU(l3,U3re,U3im);
    bool odd = ((l1+l2+l3)&1) != 0;
    for (int e=tid; e<entries; e+=blockDim.x){
      int a = e/(d2*d3); int rem = e - a*d2*d3; int b = rem/d3; int c = rem - b*d3;
      double accre=0.0, accim=0.0;
      for (int u=0;u<d1;u++){
        double are=U1re[a][u], aim=-U1im[a][u];
        if (are==0.0 && aim==0.0) continue;
        for (int v=0;v<d2;v++){
          double bre=U2re[b][v], bim=-U2im[b][v];
          if (bre==0.0 && bim==0.0) continue;
          double t1re = are*bre - aim*bim;
          double t1im = are*bim + aim*bre;
          for (int w=0;w<d3;w++){
            double cgv = cgc_dev(l1,u-l1,l2,v-l2,l3,w-l3);
            if (cgv==0.0) continue;
            double cre=U3re[c][w], cim=U3im[c][w];
            accre += (t1re*cre - t1im*cim)*cgv;
            accim += (t1re*cim + t1im*cre)*cgv;
          }
        }
      }
      // odd coupling: multiply by -i, take real part -> imag accumulator
      cg[tt*128 + e] = (float)(odd ? accim : accre);
    }
  }
}

// ---------------- invariant norms ----------------
__global__ __launch_bounds__(128) void inv_kernel(const float* __restrict__ x1d, float* __restrict__ inv){
  int n = blockIdx.x;
  for (int ch = threadIdx.x; ch < NCHN; ch += 128){
    int t = 0;
    #pragma unroll
    for (int k=1;k<6;k++) if (ch >= c_CHOFF[k]) t = k;
    int c = ch - c_CHOFF[t];
    int nt = c_NT[t], d = 2*c_LT[t]+1;
    const float* b = x1d + n*TOT + c_BOFF[t] + c;
    float s = 1e-12f;
    for (int m=0;m<d;m++){ float v = b[m*nt]; s += v*v; }
    inv[n*NCHN+ch] = sqrtf(s);
  }
}

// ---------------- batchnorm statistics (deterministic single block) ----------------
__global__ void bnstats_kernel(const float* __restrict__ inv, float* __restrict__ mean, float* __restrict__ var){
  int ch = threadIdx.x;
  if (ch >= NCHN) return;
  float s=0.f, sq=0.f;
  for (int n=0;n<NROW;n++){ float v = inv[n*NCHN+ch]; s+=v; sq+=v*v; }
  float m = s * (1.0f/NROW);
  mean[ch] = m;
  var[ch]  = sq*(1.0f/NROW) - m*m;
}

// ---------------- helpers ----------------
__device__ inline void map576(int e, int& t, int& m, int& c){
  int tt = 0;
  #pragma unroll
  for (int k=1;k<6;k++) if (e >= c_BOFF[k]) tt = k;
  t = tt;
  int loc = e - c_BOFF[tt];
  int n = c_NT[tt];
  m = loc / n;
  c = loc - m*n;
}
__device__ inline float swishf(float x){
  // x * sigmoid(x) using v_rcp_f32 (avoids IEEE divide expansion)
  return x * __builtin_amdgcn_rcpf(1.0f + __expf(-x));
}

// ---------------- one-body path: BN -> WMMA MLP -> gauge -> ielin1 -> CGP -> ielin2+res -> scale ----------------
__global__ __launch_bounds__(128) void onebody_kernel(
    const float* __restrict__ x1d, Ptr6 w1,
    const float* __restrict__ mw1, const float* __restrict__ mb1,
    const float* __restrict__ mw2, const float* __restrict__ mb2,
    const float* __restrict__ gamma, const float* __restrict__ beta,
    Ptr6 w2, const float* __restrict__ wsf, float* __restrict__ out1)
{
  __shared__ float sA[16][NCHN];   // BN out -> final nvec
  __shared__ float sB[16][NCHN];   // MLP hidden
  __shared__ float grow[TOT];
  __shared__ float g1row[TOT];
  __shared__ float zrow[ZTOT];

  const float* inv  = wsf + WS_INV;
  const float* mean = wsf + WS_MEAN;
  const float* var  = wsf + WS_VAR;
  const float* cg   = wsf + WS_CG;

  int tid  = threadIdx.x;
  int lane = tid & 31;
  int wv   = tid >> 5;
  int row0 = blockIdx.x * 16;

  // ---- RepNorm invariant content (batchnorm) ----
  for (int e = tid; e < 16*NCHN; e += 128){
    int r = e / NCHN, ch = e - r*NCHN;
    float iv = inv[(row0+r)*NCHN + ch];
    sA[r][ch] = gamma[ch]*(iv - mean[ch])*rsqrtf(var[ch] + 1e-5f) + beta[ch];
  }
  __syncthreads();

  // ---- MLP GEMM1: sB = swish(sA @ mw1 + mb1), WMMA f32 16x16x4 ----
  {
    int m  = lane & 15;
    int kh = (lane >> 4) * 2;         // K pair select per ISA A/B layout
    for (int tile = wv; tile < 14; tile += 4){
      int nch = tile*16 + (lane & 15);
      float bias = mb1[nch];
      v8f acc;
      #pragma unroll
      for (int i=0;i<8;i++) acc[i] = bias;
      for (int k0 = 0; k0 < NCHN; k0 += 4){
        v2f a, b;
        a.x = sA[m][k0+kh];   a.y = sA[m][k0+kh+1];
        b.x = mw1[(k0+kh)*NCHN + nch];
        b.y = mw1[(k0+kh+1)*NCHN + nch];
        acc = __builtin_amdgcn_wmma_f32_16x16x4_f32(false, a, false, b, (short)0, acc, false, false);
      }
      #pragma unroll
      for (int r8=0;r8<8;r8++){
        int mm = r8 + ((lane >> 4) << 3);   // C/D: VGPR r -> rows r (lanes 0-15) / r+8 (lanes 16-31)
        sB[mm][nch] = swishf(acc[r8]);
      }
    }
  }
  __syncthreads();

  // ---- MLP GEMM2: sA = sB @ mw2 + mb2 ----
  {
    int m  = lane & 15;
    int kh = (lane >> 4) * 2;
    for (int tile = wv; tile < 14; tile += 4){
      int nch = tile*16 + (lane & 15);
      float bias = mb2[nch];
      v8f acc;
      #pragma unroll
      for (int i=0;i<8;i++) acc[i] = bias;
      for (int k0 = 0; k0 < NCHN; k0 += 4){
        v2f a, b;
        a.x = sB[m][k0+kh];   a.y = sB[m][k0+kh+1];
        b.x = mw2[(k0+kh)*NCHN + nch];
        b.y = mw2[(k0+kh+1)*NCHN + nch];
        acc = __builtin_amdgcn_wmma_f32_16x16x4_f32(false, a, false, b, (short)0, acc, false, false);
      }
      #pragma unroll
      for (int r8=0;r8<8;r8++){
        int mm = r8 + ((lane >> 4) << 3);
        sA[mm][nch] = acc[r8];
      }
    }
  }
  __syncthreads();

  // ---- per-row equivariant path ----
  for (int r = 0; r < 16; ++r){
    int row = row0 + r;
    // gauge direction g = x / (inv + eps_gauge)
    for (int e = tid; e < TOT; e += 128){
      int t,m,c; map576(e,t,m,c);
      float iv = inv[row*NCHN + c_CHOFF[t] + c];
      grow[e] = x1d[row*TOT + e] / (iv + 1e-2f);
    }
    __syncthreads();
    // ielin1: block-diagonal per-irrep linear
    for (int e = tid; e < TOT; e += 128){
      int t,m,o; map576(e,t,m,o);
      int n = c_NT[t];
      const float* W = w1.p[t];
      const float* gb = &grow[c_BOFF[t] + m*n];
      float acc = 0.f;
      for (int i=0;i<n;i++) acc += gb[i]*W[i*n+o];
      g1row[e] = acc;
    }
    __syncthreads();
    // CGP: channel-wise Clebsch-Gordan self-product
    for (int wI = tid; wI < ZTOT; wI += 128){
      int s = 0;
      while (wI >= c_SEG_BASE[s+1]) s++;
      int local = wI - c_SEG_BASE[s];
      int nc = c_SEG_NC[s];
      int m = local / nc, c = local - m*nc;
      int t1 = c_SEG_T1[s], t2 = c_SEG_T2[s], to = c_SEG_TOUT[s];
      int d1 = 2*c_LT[t1]+1, d2 = 2*c_LT[t2]+1, d3 = 2*c_LT[to]+1;
      const float* C  = cg + c_SEG_TYPE[s]*128;
      const float* g1 = &g1row[c_BOFF[t1] + c];
      const float* g2 = &g1row[c_BOFF[t2] + c];
      int n1 = c_NT[t1], n2 = c_NT[t2];
      float acc = 0.f;
      for (int a=0;a<d1;a++){
        float ga = g1[a*n1];
        const float* Crow = C + (a*d2)*d3 + m;
        for (int b=0;b<d2;b++)
          acc += ga * g2[b*n2] * Crow[b*d3];
      }
      zrow[c_ZBASE[to] + m*c_METAOUT[to] + c_SEG_CHOFF[s] + c] = acc;
    }
    __syncthreads();
    // ielin2 + residual + scalar_mul by nvec, store x_1d_new
    for (int e = tid; e < TOT; e += 128){
      int t,m,o; map576(e,t,m,o);
      int K = c_METAOUT[t], n = c_NT[t];
      const float* W  = w2.p[t];
      const float* zp = &zrow[c_ZBASE[t] + m*K];
      float acc = 0.f;
      for (int i=0;i<K;i++) acc += zp[i]*W[i*n+o];
      float res = acc + g1row[e];
      out1[row*TOT + e] = res * sA[r][c_CHOFF[t] + o];
    }
    __syncthreads();
  }
}

// ---------------- twobody: a_raw[n,c] = fold^T . x2d . fold, then fold gate into 4x4 matrix M[n] ----------------
__global__ __launch_bounds__(256) void araw_kernel(const float* __restrict__ x2d,
                                                   const float* __restrict__ out1,
                                                   const float* __restrict__ gw,
                                                   const float* __restrict__ gb,
                                                   const float* __restrict__ lin,
                                                   float* __restrict__ Mout){
  __shared__ float sfold[TOT];
  __shared__ float4 sred[256];
  int n = blockIdx.x, tid = threadIdx.x;
  for (int e=tid;e<TOT;e+=256) sfold[e] = out1[n*TOT+e];
  __syncthreads();
  const float4* x4 = (const float4*)x2d + (size_t)n * (DD*DD);
  float ax=0.f, ay=0.f, az=0.f, aw=0.f;
  for (int e=tid; e<DD*DD; e+=256){
    int i = e / DD, j = e - i*DD;
    float4 v = x4[e];
    const float* fj = &sfold[j*4];
    const float* fi = &sfold[i*4];
    ax += v.x * fj[0]*fi[0];
    ay += v.y * fj[1]*fi[1];
    az += v.z * fj[2]*fi[2];
    aw += v.w * fj[3]*fi[3];
  }
  sred[tid] = make_float4(ax,ay,az,aw);
  __syncthreads();
  for (int s=128;s>0;s>>=1){
    if (tid<s){
      float4 a=sred[tid], b=sred[tid+s];
      sred[tid]=make_float4(a.x+b.x, a.y+b.y, a.z+b.z, a.w+b.w);
    }
    __syncthreads();
  }
  // M[n][c][o] = tb_lin_w[c][o] * swish(araw @ tb_gate_w + tb_gate_b)[o]
  if (tid < 16){
    int c = tid >> 2, o = tid & 3;
    float4 a = sred[0];
    float ar[4] = {a.x, a.y, a.z, a.w};
    float s = ar[0]*gw[0*4+o] + ar[1]*gw[1*4+o] + ar[2]*gw[2*4+o] + ar[3]*gw[3*4+o] + gb[o];
    float gate = swishf(s);
    Mout[n*16 + c*4 + o] = lin[c*4+o] * gate;
  }
}

// ---------------- twobody streaming pass: out2[n,i,j,:] = x2d[n,i,j,:] @ M[n] ----------------
// Grid: NROW * BLKS_PER_ROW blocks of 256 threads; exactly one float4 per thread, no divides in hot path.
__global__ __launch_bounds__(256) void x2dnew_kernel(const float* __restrict__ x2d,
                                                     const float* __restrict__ Mmat,
                                                     float* __restrict__ out2){
  int blk = blockIdx.x;
  int n   = blk / BLKS_PER_ROW;
  int p   = (blk - n*BLKS_PER_ROW)*256 + threadIdx.x;   // 0..20735 (exact)
  int idx = n*(DD*DD) + p;

  const float4* Mld = (const float4*)(Mmat + n*16);
  float4 m0 = Mld[0], m1 = Mld[1], m2 = Mld[2], m3 = Mld[3];

  float4 v = ((const float4*)x2d)[idx];
  float4 ov;
  ov.x = v.x*m0.x + v.y*m1.x + v.z*m2.x + v.w*m3.x;
  ov.y = v.x*m0.y + v.y*m1.y + v.z*m2.y + v.w*m3.y;
  ov.z = v.x*m0.z + v.y*m1.z + v.z*m2.z + v.w*m3.z;
  ov.w = v.x*m0.w + v.y*m1.w + v.z*m2.w + v.w*m3.w;
  ((float4*)out2)[idx] = ov;
}

extern "C" void kernel_launch(void* const* d_in, const int* in_sizes, int n_in,
                              void* d_out, int out_size, void* d_ws, size_t ws_size,
                              hipStream_t stream) {
  const float *x1d, *x2d, *mw1, *mb1, *mw2, *mb2, *gamma, *beta, *lin, *gw, *gb;
  Ptr6 w1, w2;
  static const int W1OFF[6] = {0,4096,5120,7424,8448,9472};        // 64^2,32^2,48^2,32^2,32^2,16^2
  static const int W2OFF[6] = {0,14336,16896,29184,34304,41472};   // 224*64,80*32,256*48,160*32,224*32,144*16

  if (n_in >= 23) {
    // tuples flattened as separate inputs
    x1d=(const float*)d_in[0]; x2d=(const float*)d_in[1];
    for (int i=0;i<6;i++) w1.p[i]=(const float*)d_in[2+i];
    mw1=(const float*)d_in[8];  mb1=(const float*)d_in[9];
    mw2=(const float*)d_in[10]; mb2=(const float*)d_in[11];
    gamma=(const float*)d_in[12]; beta=(const float*)d_in[13];
    for (int i=0;i<6;i++) w2.p[i]=(const float*)d_in[14+i];
    lin=(const float*)d_in[20]; gw=(const float*)d_in[21]; gb=(const float*)d_in[22];
  } else {
    // tuples packed as single concatenated buffers
    x1d=(const float*)d_in[0]; x2d=(const float*)d_in[1];
    const float* w1c=(const float*)d_in[2];
    for (int i=0;i<6;i++) w1.p[i]=w1c+W1OFF[i];
    mw1=(const float*)d_in[3]; mb1=(const float*)d_in[4];
    mw2=(const float*)d_in[5]; mb2=(const float*)d_in[6];
    gamma=(const float*)d_in[7]; beta=(const float*)d_in[8];
    const float* w2c=(const float*)d_in[9];
    for (int i=0;i<6;i++) w2.p[i]=w2c+W2OFF[i];
    lin=(const float*)d_in[10]; gw=(const float*)d_in[11]; gb=(const float*)d_in[12];
  }

  float* wsf  = (float*)d_ws;
  float* out1 = (float*)d_out;             // x_1d_new: 512*576
  float* out2 = out1 + NROW*TOT;           // x_2d_new: 512*144*144*4

  cg_init_kernel<<<1, 128, 0, stream>>>(wsf + WS_CG);
  inv_kernel<<<NROW, 128, 0, stream>>>(x1d, wsf + WS_INV);
  bnstats_kernel<<<1, 224, 0, stream>>>(wsf + WS_INV, wsf + WS_MEAN, wsf + WS_VAR);
  onebody_kernel<<<NROW/16, 128, 0, stream>>>(x1d, w1, mw1, mb1, mw2, mb2,
                                              gamma, beta, w2, wsf, out1);
  araw_kernel<<<NROW, 256, 0, stream>>>(x2d, out1, gw, gb, lin, wsf + WS_M);
  x2dnew_kernel<<<NROW*BLKS_PER_ROW, 256, 0, stream>>>(x2d, wsf + WS_M, out2);
}